// Model_3178275799146
// MI455X (gfx1250) — compile-verified
//
#include <hip/hip_runtime.h>
#include <math.h>

typedef __attribute__((ext_vector_type(2))) float v2f;
typedef __attribute__((ext_vector_type(4))) float v4f;
typedef __attribute__((ext_vector_type(8))) float v8f;

#define NNODES 50000
#define NEDGES 800000
#define FIN    128
#define FHID   256
#define FOUT   128

// ---------------------------------------------------------------------------
// Degree / normalization kernels
// ---------------------------------------------------------------------------
__global__ void init_deg_kernel(float* __restrict__ deg) {
    int i = blockIdx.x * blockDim.x + threadIdx.x;
    if (i < NNODES) deg[i] = 1.0f;   // self-loop contributes 1 to every node
}

__global__ void count_deg_kernel(const int* __restrict__ dst, float* __restrict__ deg) {
    int e = blockIdx.x * blockDim.x + threadIdx.x;
    if (e < NEDGES) atomicAdd(&deg[dst[e]], 1.0f);
}

__global__ void rsqrt_deg_kernel(float* __restrict__ deg) {
    int i = blockIdx.x * blockDim.x + threadIdx.x;
    if (i < NNODES) deg[i] = rsqrtf(deg[i]);   // deg >= 1 always (self-loop)
}

// ---------------------------------------------------------------------------
// FP32 WMMA GEMM: C[M x NB] = A[M x K] * B[K x NB]
// One wave32 computes one 16x16 tile of C using V_WMMA_F32_16X16X4_F32.
//
// ISA 7.12.2 layouts (wave32):
//   A 16x4 : lane L holds M=(L&15); VGPR0={K=0|K=2 by lane half}, VGPR1={K=1|K=3}
//            -> per lane a contiguous float2 at row (L&15), col k + 2*(L>>4)
//   B 4x16 : lane L holds N=(L&15); VGPR0 row K=0 (lanes 0-15) / K=2 (16-31),
//            VGPR1 row K=1 / K=3
//   C/D    : 8 VGPRs, VGPR v = element (M = v + 8*(L>>4), N = L&15)
// ---------------------------------------------------------------------------
template <int K, int NB>
__global__ __launch_bounds__(32) void gemm_wmma_f32_kernel(
    const float* __restrict__ A, const float* __restrict__ B,
    float* __restrict__ C)
{
    const int lane  = threadIdx.x & 31;
    const int m0    = blockIdx.x * 16;
    const int n0    = blockIdx.y * 16;
    const int ml    = lane & 15;        // row within tile (A) / col within tile (B,C)
    const int khalf = (lane >> 4) * 2;  // 0 for lanes 0-15, 2 for lanes 16-31

    const float* __restrict__ arow = A + (size_t)(m0 + ml) * K;
    const float* __restrict__ bcol = B + (size_t)(n0 + ml);

    v8f acc = {};
    #pragma unroll 4
    for (int k = 0; k < K; k += 4) {
        v2f a;
        a.x = arow[k + khalf + 0];
        a.y = arow[k + khalf + 1];
        v2f b;
        b.x = bcol[(size_t)(k + khalf + 0) * NB];
        b.y = bcol[(size_t)(k + khalf + 1) * NB];
        acc = __builtin_amdgcn_wmma_f32_16x16x4_f32(
            /*neg_a=*/false, a, /*neg_b=*/false, b,
            /*c_mod=*/(short)0, acc, /*reuse_a=*/false, /*reuse_b=*/false);
    }

    const int mhalf = (lane >> 4) * 8;
    #pragma unroll
    for (int v = 0; v < 8; ++v) {
        C[(size_t)(m0 + mhalf + v) * NB + (n0 + ml)] = acc[v];
    }
}

// ---------------------------------------------------------------------------
// Edge scatter: out[dst] += h[src] * (dinv[src]*dinv[dst]).
// One wave per edge; lane l moves 4 consecutive floats per 128-wide pass
// (float4 coalesced load + 4 global_atomic_add_f32).
// ---------------------------------------------------------------------------
template <int F>
__global__ __launch_bounds__(256) void scatter_edges_kernel(
    const int* __restrict__ src, const int* __restrict__ dst,
    const float* __restrict__ dinv, const float* __restrict__ h,
    float* __restrict__ out)
{
    const int wave = blockIdx.x * (blockDim.x >> 5) + (threadIdx.x >> 5);
    const int lane = threadIdx.x & 31;
    if (wave >= NEDGES) return;

    const int s = src[wave];
    const int d = dst[wave];
    const float w = dinv[s] * dinv[d];

    const float* __restrict__ hs = h + (size_t)s * F;
    float* __restrict__ od = out + (size_t)d * F;

    #pragma unroll
    for (int j = 0; j < F; j += 128) {
        const int f = j + lane * 4;
        v4f val = *(const v4f*)(hs + f);
        atomicAdd(od + f + 0, val.x * w);
        atomicAdd(od + f + 1, val.y * w);
        atomicAdd(od + f + 2, val.z * w);
        atomicAdd(od + f + 3, val.w * w);
    }
}

// ---------------------------------------------------------------------------
// Epilogue: out = relu(agg + dinv[i]^2 * h + bias)   (self-loop folded in)
// One thread per element. Safe for agg == out (in-place).
// ---------------------------------------------------------------------------
template <int F>
__global__ __launch_bounds__(256) void epilogue_kernel(
    const float* __restrict__ agg, const float* __restrict__ h,
    const float* __restrict__ dinv, const float* __restrict__ bias,
    float* __restrict__ out)
{
    const size_t idx = (size_t)blockIdx.x * blockDim.x + threadIdx.x;
    if (idx >= (size_t)NNODES * F) return;
    const int node = (int)(idx / F);
    const int f    = (int)(idx % F);
    const float di = dinv[node];
    const float v  = agg[idx] + di * di * h[idx] + bias[f];
    out[idx] = fmaxf(v, 0.0f);
}

// ---------------------------------------------------------------------------
// Launcher
// ---------------------------------------------------------------------------
extern "C" void kernel_launch(void* const* d_in, const int* in_sizes, int n_in,
                              void* d_out, int out_size, void* d_ws, size_t ws_size,
                              hipStream_t stream) {
    const float* x   = (const float*)d_in[0];
    const int*   ei  = (const int*)d_in[1];          // [2, E] flattened
    const float* W1  = (const float*)d_in[2];
    const float* b1  = (const float*)d_in[3];
    const float* W2  = (const float*)d_in[4];
    const float* b2  = (const float*)d_in[5];
    float*       out = (float*)d_out;                // [N, FOUT]

    const int* src = ei;
    const int* dst = ei + NEDGES;

    // Workspace carve-up (floats)
    float* dinv = (float*)d_ws;                      // N            (deg, then dinv)
    float* h1   = dinv + NNODES;                     // N*FHID       (h1, later reused as h2)
    float* agg1 = h1 + (size_t)NNODES * FHID;        // N*FHID       (layer-1 accumulator)
    float* h2   = h1;                                // reuse: h1 dead after epilogue1

    // Re-zero accumulation targets every call (graph-replay safe)
    hipMemsetAsync(agg1, 0, (size_t)NNODES * FHID * sizeof(float), stream);
    hipMemsetAsync(out,  0, (size_t)NNODES * FOUT * sizeof(float), stream);

    // Degree -> dinv
    init_deg_kernel<<<(NNODES + 255) / 256, 256, 0, stream>>>(dinv);
    count_deg_kernel<<<(NEDGES + 255) / 256, 256, 0, stream>>>(dst, dinv);
    rsqrt_deg_kernel<<<(NNODES + 255) / 256, 256, 0, stream>>>(dinv);

    // Layer 1: h1 = x @ W1  (50000x128 @ 128x256)
    {
        dim3 grid(NNODES / 16, FHID / 16);
        gemm_wmma_f32_kernel<FIN, FHID><<<grid, 32, 0, stream>>>(x, W1, h1);
    }
    // agg1 += h1[src] * norm over edges
    scatter_edges_kernel<FHID><<<(NEDGES + 7) / 8, 256, 0, stream>>>(src, dst, dinv, h1, agg1);
    // agg1 = relu(agg1 + dinv^2 * h1 + b1)   (in place)
    epilogue_kernel<FHID><<<((size_t)NNODES * FHID + 255) / 256, 256, 0, stream>>>(
        agg1, h1, dinv, b1, agg1);

    // Layer 2: h2 = agg1 @ W2  (50000x256 @ 256x128)
    {
        dim3 grid(NNODES / 16, FOUT / 16);
        gemm_wmma_f32_kernel<FHID, FOUT><<<grid, 32, 0, stream>>>(agg1, W2, h2);
    }
    // out += h2[src] * norm over edges
    scatter_edges_kernel<FOUT><<<(NEDGES + 7) / 8, 256, 0, stream>>>(src, dst, dinv, h2, out);
    // out = relu(out + dinv^2 * h2 + b2)   (in place)
    epilogue_kernel<FOUT><<<((size_t)NNODES * FOUT + 255) / 256, 256, 0, stream>>>(
        out, h2, dinv, b2, out);
}